// PatchlenWiseEmbedding_17274358465002
// MI455X (gfx1250) — compile-verified
//
#include <hip/hip_runtime.h>
#include <hip/hip_bf16.h>

typedef __attribute__((ext_vector_type(2))) float v2f;
typedef __attribute__((ext_vector_type(8))) float v8f;

#define D_MODEL 512
#define TOKENS  1608
#define L_SEQ   8192
#define NVARS   8

// -ln(10000)/512
#define PE_COEF (-0.017988946039325602f)

// token index of the global token for each variable: tok_off[v] + n_p[v]
__constant__ int c_glb_tok[NVARS] = {512, 1025, 1154, 1283, 1412, 1541, 1574, 1607};

// One workgroup = 256 threads = 8 wave32.
// Computes a 16-patch x 128-column tile of the embedding for one (b, variable).
//   A (16 x PL)  : patch tile, staged through LDS (handles SAMPLING stride SF)
//   B (PL x 16)  : w^T fragment per wave (w is [512, PL] row-major)
//   D (16 x 16)  : f32 accumulator via V_WMMA_F32_16X16X4_F32, K unrolled PL/4 times
template<int PL, int SF>
__global__ __launch_bounds__(256) void patch_embed_wmma(
    const float* __restrict__ xv,    // x + v*L ; batch stride NVARS*L
    const float* __restrict__ w,     // [D_MODEL, PL]
    const float* __restrict__ bias,  // [D_MODEL]
    const float* __restrict__ chv,   // channel_emb + v*D_MODEL
    float* __restrict__ out,         // [B, TOKENS, D_MODEL]
    int tok_off)
{
    __shared__ float sA[16 * (PL + 4)];   // +4 floats pad: kills LDS bank conflicts, keeps 8B align

    const int tid = threadIdx.x;
    const int nt  = blockIdx.x;           // 0..3 : 128-column block of d_model
    const int mt  = blockIdx.y;           // patch tile index
    const int b   = blockIdx.z;           // batch

    // ---- stage 16 x PL patch tile (strided gather, element j at stride SF) ----
    const float* xb = xv + (size_t)b * (NVARS * L_SEQ) + (size_t)mt * 16 * PL * SF;
    #pragma unroll
    for (int t = tid; t < 16 * PL; t += 256) {
        const int m = t / PL;
        const int k = t % PL;
        sA[m * (PL + 4) + k] = xb[(m * PL + k) * SF];
    }
    __syncthreads();

    // ---- WMMA fragments (ISA 7.12.2 f32 layouts) ----
    const int lane = tid & 31;
    const int wv   = tid >> 5;
    const int n0   = nt * 128 + wv * 16;     // this wave's 16 columns of d_model
    const int mr   = lane & 15;              // A row (M) / B column (N)
    const int kh   = (lane >> 4) << 1;       // K sub-pair: lanes 0-15 -> K0,K1 ; 16-31 -> K2,K3

    const float* ap = sA + mr * (PL + 4) + kh;
    const float* bp = w + (size_t)(n0 + mr) * PL + kh;   // w^T[k][n] = w[n*PL + k]

    v8f acc = {0.f, 0.f, 0.f, 0.f, 0.f, 0.f, 0.f, 0.f};
    #pragma unroll
    for (int kb = 0; kb < PL; kb += 4) {
        v2f a  = *(const v2f*)(ap + kb);
        v2f bm = *(const v2f*)(bp + kb);
        acc = __builtin_amdgcn_wmma_f32_16x16x4_f32(
            /*neg_a=*/false, a, /*neg_b=*/false, bm,
            /*c_mod=*/(short)0, acc, /*reuse_a=*/false, /*reuse_b=*/false);
    }

    // ---- fused epilogue: + bias + channel_emb + sinusoidal pos emb, single store ----
    const int   col      = n0 + mr;                       // C/D: N = lane&15 (per 16-col tile)
    const float base_add = bias[col] + chv[col];
    const float dexp     = __expf((float)(col & ~1) * PE_COEF);
    const int   rb       = (lane >> 4) << 3;              // lanes 16-31 hold M = r+8
    float* orow = out + ((size_t)b * TOKENS + tok_off + mt * 16) * D_MODEL + col;
    #pragma unroll
    for (int r = 0; r < 8; ++r) {
        const int   m   = rb + r;
        const float arg = (float)(mt * 16 + m) * dexp;
        const float pe  = (col & 1) ? __cosf(arg) : __sinf(arg);
        orow[(size_t)m * D_MODEL] = acc[r] + base_add + pe;
    }
}

// Fill the 8 (vars) x 32 (batch) global-token rows: glb + channel_emb[v]
__global__ __launch_bounds__(256) void global_token_kernel(
    const float* __restrict__ glb,     // [D_MODEL]
    const float* __restrict__ chemb,   // [NVARS, D_MODEL]
    float* __restrict__ out)
{
    const int idx  = blockIdx.x * 256 + threadIdx.x;   // NVARS*32*D_MODEL total
    const int col  = idx & (D_MODEL - 1);
    const int rest = idx >> 9;
    const int b    = rest & 31;
    const int v    = rest >> 5;
    out[((size_t)b * TOKENS + c_glb_tok[v]) * D_MODEL + col] = glb[col] + chemb[v * D_MODEL + col];
}

extern "C" void kernel_launch(void* const* d_in, const int* in_sizes, int n_in,
                              void* d_out, int out_size, void* d_ws, size_t ws_size,
                              hipStream_t stream) {
    const float* x    = (const float*)d_in[0];
    const float* w16  = (const float*)d_in[1];  const float* b16  = (const float*)d_in[2];
    const float* w32  = (const float*)d_in[3];  const float* b32  = (const float*)d_in[4];
    const float* w64  = (const float*)d_in[5];  const float* b64  = (const float*)d_in[6];
    const float* w128 = (const float*)d_in[7];  const float* b128 = (const float*)d_in[8];
    const float* glb  = (const float*)d_in[9];
    const float* ce   = (const float*)d_in[10];
    float* out = (float*)d_out;

    const dim3 blk(256);
    // grid = (4 column-blocks of 128, n_p/16 patch tiles, B=32)
    patch_embed_wmma<16, 1><<<dim3(4, 32, 32), blk, 0, stream>>>(x + 0 * L_SEQ, w16,  b16,  ce + 0 * D_MODEL, out, 0);
    patch_embed_wmma<16, 1><<<dim3(4, 32, 32), blk, 0, stream>>>(x + 1 * L_SEQ, w16,  b16,  ce + 1 * D_MODEL, out, 513);
    patch_embed_wmma<32, 2><<<dim3(4,  8, 32), blk, 0, stream>>>(x + 2 * L_SEQ, w32,  b32,  ce + 2 * D_MODEL, out, 1026);
    patch_embed_wmma<32, 2><<<dim3(4,  8, 32), blk, 0, stream>>>(x + 3 * L_SEQ, w32,  b32,  ce + 3 * D_MODEL, out, 1155);
    patch_embed_wmma<64, 1><<<dim3(4,  8, 32), blk, 0, stream>>>(x + 4 * L_SEQ, w64,  b64,  ce + 4 * D_MODEL, out, 1284);
    patch_embed_wmma<64, 1><<<dim3(4,  8, 32), blk, 0, stream>>>(x + 5 * L_SEQ, w64,  b64,  ce + 5 * D_MODEL, out, 1413);
    patch_embed_wmma<128,2><<<dim3(4,  2, 32), blk, 0, stream>>>(x + 6 * L_SEQ, w128, b128, ce + 6 * D_MODEL, out, 1542);
    patch_embed_wmma<128,2><<<dim3(4,  2, 32), blk, 0, stream>>>(x + 7 * L_SEQ, w128, b128, ce + 7 * D_MODEL, out, 1575);

    global_token_kernel<<<dim3(NVARS * 32 * D_MODEL / 256), blk, 0, stream>>>(glb, ce, out);
}